// HeteroTemporalGATLayer_38749194944550
// MI455X (gfx1250) — compile-verified
//
#include <hip/hip_runtime.h>
#include <hip/hip_bf16.h>

#define NN 50000
#define EE 500000
#define HIDD 128
#define HH 4
#define DH 32
#define TDD 32

typedef __attribute__((ext_vector_type(2))) float v2f;
typedef __attribute__((ext_vector_type(8))) float v8f;
typedef __attribute__((ext_vector_type(4))) unsigned v4u;
typedef __attribute__((ext_vector_type(4))) int v4i;
typedef __attribute__((ext_vector_type(8))) int v8i;

__device__ __forceinline__ float leaky_f(float z) { return z >= 0.f ? z : 0.2f * z; }

// ---------------- fill kernels ----------------
__global__ void fill_f32_kernel(float* __restrict__ p, float v, int n) {
  int i = blockIdx.x * blockDim.x + threadIdx.x;
  if (i < n) p[i] = v;
}
__global__ void fill_i32_kernel(int* __restrict__ p, int v, int n) {
  int i = blockIdx.x * blockDim.x + threadIdx.x;
  if (i < n) p[i] = v;
}

// ---------------- TDM: stage 64KB B matrix (128x128 f32) into LDS ----------------
// 1D D# per CDNA5 ISA: 16384 elements, data_size=4B, type=2.
__device__ __forceinline__ void tdm_load_B_to_lds(const float* B, float* sB) {
  unsigned long long ga = (unsigned long long)B;
  unsigned lds_off = (unsigned)(unsigned long long)(void*)sB;
  const unsigned nelem = HIDD * HIDD;  // 16384

  v4u g0;
  g0.x = 1u;                                        // count=1, user-mode descriptor
  g0.y = lds_off;                                   // lds_addr [63:32]
  g0.z = (unsigned)(ga & 0xFFFFFFFFu);              // global_addr [95:64]
  g0.w = (unsigned)((ga >> 32) & 0x01FFFFFFu)       // global_addr [120:96]
         | (2u << 30);                              // type=2 ("image") [127:126]

  v8i g1;
  g1[0] = (int)(2u << 16);                          // workgroup_mask=0, data_size=2 (4B)
  g1[1] = (int)((nelem & 0xFFFFu) << 16);           // tensor_dim0[15:0] at bits 63:48
  g1[2] = (int)((nelem >> 16) | (1u << 16));        // tensor_dim0[31:16], tensor_dim1=1
  g1[3] = (int)(nelem << 16);                       // tensor_dim1 hi=0, tile_dim0=16384
  g1[4] = 1;                                        // tile_dim1=1, tile_dim2=0
  g1[5] = (int)nelem;                               // tensor_dim0_stride[31:0]
  g1[6] = 0;                                        // stride hi, tensor_dim1_stride lo
  g1[7] = 0;

  v4i g2 = {0, 0, 0, 0};
  v4i g3 = {0, 0, 0, 0};
#if defined(__clang_major__) && (__clang_major__ >= 23)
  v8i g4 = {0, 0, 0, 0, 0, 0, 0, 0};
  __builtin_amdgcn_tensor_load_to_lds(g0, g1, g2, g3, g4, 0);
#else
  __builtin_amdgcn_tensor_load_to_lds(g0, g1, g2, g3, 0);
#endif
  __builtin_amdgcn_s_wait_tensorcnt(0);
}

// ---------------- f32 WMMA GEMM (K = Nc = 128): C = A[M x 128] @ B[128 x 128] -----
// B staged in LDS via TDM (one wave issues). Each wave owns a 16x128 output strip:
// 8 accumulator tiles, so each per-lane A float2 load feeds 8 V_WMMA_F32_16X16X4_F32.
__global__ void __launch_bounds__(256) gemm128_wmma_kernel(
    const float* __restrict__ A, const float* __restrict__ B,
    float* __restrict__ C, int M) {
  __shared__ float sB[HIDD * HIDD];  // 64 KB of the 320 KB WGP LDS

  if (threadIdx.x < 32) {            // wave 0 issues the TDM DMA, waits TENSORcnt
    tdm_load_B_to_lds(B, sB);
  }
  __syncthreads();

  int waveInBlk = threadIdx.x >> 5;
  int lane = threadIdx.x & 31;
  int mt = blockIdx.x * 8 + waveInBlk;       // 16-row tile index
  if (mt >= (M >> 4)) return;                // uniform per wave: EXEC stays all-ones
  int m0 = mt << 4;
  int half = lane >> 4;                      // 0: lanes 0-15, 1: lanes 16-31
  int l15 = lane & 15;
  int koff = half << 1;

  v8f c[8] = {};
  const float* arow = A + (size_t)(m0 + l15) * HIDD;  // A row M = lane&15
  for (int kb = 0; kb < HIDD; kb += 4) {
    v2f a = *(const v2f*)(arow + kb + koff);          // A[M][kb+koff .. +1]
    const float* b0 = sB + (kb + koff) * HIDD + l15;
    const float* b1 = b0 + HIDD;
#pragma unroll
    for (int nt = 0; nt < 8; ++nt) {
      v2f b;
      b.x = b0[nt * 16];                              // B[kb+koff    ][nt*16 + l15]
      b.y = b1[nt * 16];                              // B[kb+koff + 1][nt*16 + l15]
      c[nt] = __builtin_amdgcn_wmma_f32_16x16x4_f32(false, a, false, b, (short)0,
                                                    c[nt], false, false);
    }
  }
  // C VGPR r of tile nt: row m0 + r + 8*half, col nt*16 + (lane&15)
  float* crow = C + (size_t)(m0 + (half << 3)) * HIDD + l15;
#pragma unroll
  for (int nt = 0; nt < 8; ++nt)
#pragma unroll
    for (int r = 0; r < 8; ++r) crow[(size_t)r * HIDD + nt * 16] = c[nt][r];
}

// ---------------- per-node attention scores ----------------
// score[n][h] = sum_d leaky(h[n][h*32+d]) * att[h][attOff + d]
__global__ void node_scores_kernel(const float* __restrict__ hmat,
                                   const float* __restrict__ att, int attOff,
                                   float* __restrict__ score, int n) {
  int i = blockIdx.x * blockDim.x + threadIdx.x;
  if (i >= n * HH) return;
  int node = i >> 2, hd = i & 3;
  const float* hp = hmat + (size_t)node * HIDD + hd * DH;
  const float* ap = att + hd * (3 * DH) + attOff;
  float s = 0.f;
#pragma unroll 8
  for (int d = 0; d < DH; ++d) s += leaky_f(hp[d]) * ap[d];
  score[i] = s;
}

// monotone int key for float atomic-max
__device__ __forceinline__ int f2key(float f) {
  int b = __float_as_int(f);
  return b >= 0 ? b : (b ^ 0x7FFFFFFF);
}
__device__ __forceinline__ float key2f(int k) {
  return __int_as_float(k >= 0 ? k : (k ^ 0x7FFFFFFF));
}

// ---------------- edge pass 1: logits + segment max ----------------
__global__ void __launch_bounds__(256) edge_logit_kernel(
    const int* __restrict__ srcIdx, const int* __restrict__ dstIdx,
    const float* __restrict__ tf, const float* __restrict__ Wt,
    const float* __restrict__ att, const float* __restrict__ score_s,
    const float* __restrict__ score_d, float* __restrict__ logits,
    int* __restrict__ mkey, int nE) {
  __shared__ float sWt[TDD * HIDD];  // 16 KB
  for (int j = threadIdx.x; j < TDD * HIDD; j += blockDim.x) sWt[j] = Wt[j];
  __syncthreads();

  int i = blockIdx.x * blockDim.x + threadIdx.x;
  if (i >= nE * HH) return;
  int e = i >> 2, hd = i & 3;

  float tr[TDD];
  const float* tp = tf + (size_t)e * TDD;
#pragma unroll
  for (int k = 0; k < TDD; ++k) tr[k] = tp[k];

  const float* at = att + hd * (3 * DH) + 2 * DH;
  float st = 0.f;
  for (int d = 0; d < DH; ++d) {
    float acc = 0.f;
#pragma unroll
    for (int k = 0; k < TDD; ++k) acc += tr[k] * sWt[k * HIDD + hd * DH + d];
    st += leaky_f(acc) * at[d];
  }
  int s = srcIdx[e], dd = dstIdx[e];
  float logit = score_s[s * HH + hd] + score_d[dd * HH + hd] + st;
  logits[i] = logit;
  atomicMax(mkey + dd * HH + hd, f2key(logit));
}

// ---------------- edge pass 2: exp + segment sum ----------------
__global__ void edge_exp_kernel(const int* __restrict__ dstIdx,
                                float* __restrict__ logits,
                                const int* __restrict__ mkey,
                                float* __restrict__ ssum, int nE) {
  int i = blockIdx.x * blockDim.x + threadIdx.x;
  if (i >= nE * HH) return;
  int e = i >> 2, hd = i & 3;
  int dd = dstIdx[e];
  float m = key2f(mkey[dd * HH + hd]);
  float ex = expf(logits[i] - m);
  logits[i] = ex;  // store ex for message pass
  atomicAdd(ssum + dd * HH + hd, ex);
}

// ---------------- edge pass 3: messages + scatter-add ----------------
__global__ void __launch_bounds__(256) edge_msg_kernel(
    const int* __restrict__ srcIdx, const int* __restrict__ dstIdx,
    const float* __restrict__ tf, const float* __restrict__ Wt,
    const float* __restrict__ vsrc, const float* __restrict__ exbuf,
    const float* __restrict__ ssum, float* __restrict__ agg, int nE) {
  __shared__ float sWt[TDD * HIDD];
  for (int j = threadIdx.x; j < TDD * HIDD; j += blockDim.x) sWt[j] = Wt[j];
  __syncthreads();

  int i = blockIdx.x * blockDim.x + threadIdx.x;  // nE*128 = 64M < 2^31
  if (i >= nE * HIDD) return;
  int e = i >> 7, d = i & (HIDD - 1), hd = d >> 5;
  int s = srcIdx[e], dd = dstIdx[e];
  float alpha = exbuf[e * HH + hd] / (ssum[dd * HH + hd] + 1e-16f);
  const float* tp = tf + (size_t)e * TDD;
  float acc = 0.f;
#pragma unroll
  for (int k = 0; k < TDD; ++k) acc += tp[k] * sWt[k * HIDD + d];
  float msg = (vsrc[(size_t)s * HIDD + d] + acc) * alpha;
  atomicAdd(agg + (size_t)dd * HIDD + d, msg);
}

// ---------------- residual + bias + LayerNorm (one wave32 per node) ----------------
__global__ void __launch_bounds__(256) ln_kernel(
    const float* __restrict__ hres, const float* __restrict__ proj,
    const float* __restrict__ bout, const float* __restrict__ lnw,
    const float* __restrict__ lnb, float* __restrict__ out, int n) {
  int wave = (blockIdx.x * blockDim.x + threadIdx.x) >> 5;
  int lane = threadIdx.x & 31;
  if (wave >= n) return;
  const float* hp = hres + (size_t)wave * HIDD;
  const float* pp = proj + (size_t)wave * HIDD;
  float x[4];
#pragma unroll
  for (int i = 0; i < 4; ++i) {
    int j = lane + 32 * i;
    x[i] = hp[j] + pp[j] + bout[j];
  }
  float sum = x[0] + x[1] + x[2] + x[3];
  for (int off = 16; off > 0; off >>= 1) sum += __shfl_xor(sum, off, 32);
  float mu = sum * (1.f / 128.f);
  float sq = 0.f;
#pragma unroll
  for (int i = 0; i < 4; ++i) { float dv = x[i] - mu; sq += dv * dv; }
  for (int off = 16; off > 0; off >>= 1) sq += __shfl_xor(sq, off, 32);
  float rs = rsqrtf(sq * (1.f / 128.f) + 1e-5f);
#pragma unroll
  for (int i = 0; i < 4; ++i) {
    int j = lane + 32 * i;
    out[(size_t)wave * HIDD + j] = (x[i] - mu) * rs * lnw[j] + lnb[j];
  }
}

// ---------------- host-side direction driver ----------------
static void run_direction(const float* x_src, const float* x_dst, const int* ei,
                          const float* tf, const float* Ws, const float* Wd,
                          const float* Wt, const float* Wv, const float* att,
                          float* agg, float* h_s, float* h_d, float* v_s,
                          float* score_s, float* score_d, int* mkey, float* ssum,
                          float* exbuf, hipStream_t stream) {
  const int rowTiles = NN / 16;                     // 3125
  const int gemmBlocks = (rowTiles + 7) / 8;        // 391
  gemm128_wmma_kernel<<<gemmBlocks, 256, 0, stream>>>(x_src, Ws, h_s, NN);
  gemm128_wmma_kernel<<<gemmBlocks, 256, 0, stream>>>(x_dst, Wd, h_d, NN);
  gemm128_wmma_kernel<<<gemmBlocks, 256, 0, stream>>>(x_src, Wv, v_s, NN);

  int nThr = NN * HH;
  node_scores_kernel<<<(nThr + 255) / 256, 256, 0, stream>>>(h_s, att, 0, score_s, NN);
  node_scores_kernel<<<(nThr + 255) / 256, 256, 0, stream>>>(h_d, att, DH, score_d, NN);

  fill_i32_kernel<<<(nThr + 255) / 256, 256, 0, stream>>>(mkey, (int)0x80000000, nThr);
  fill_f32_kernel<<<(nThr + 255) / 256, 256, 0, stream>>>(ssum, 0.f, nThr);

  int eThr = EE * HH;
  edge_logit_kernel<<<(eThr + 255) / 256, 256, 0, stream>>>(
      ei, ei + EE, tf, Wt, att, score_s, score_d, exbuf, mkey, EE);
  edge_exp_kernel<<<(eThr + 255) / 256, 256, 0, stream>>>(ei + EE, exbuf, mkey, ssum, EE);
  edge_msg_kernel<<<(EE * HIDD) / 256, 256, 0, stream>>>(
      ei, ei + EE, tf, Wt, v_s, exbuf, ssum, agg, EE);
}

extern "C" void kernel_launch(void* const* d_in, const int* in_sizes, int n_in,
                              void* d_out, int out_size, void* d_ws, size_t ws_size,
                              hipStream_t stream) {
  const float* h_user    = (const float*)d_in[0];
  const float* h_item    = (const float*)d_in[1];
  const int*   ei_ui     = (const int*)d_in[2];
  const float* t_ui      = (const float*)d_in[3];
  const int*   ei_iu     = (const int*)d_in[4];
  const float* t_iu      = (const float*)d_in[5];
  const float* W_src_ui  = (const float*)d_in[6];
  const float* W_dst_ui  = (const float*)d_in[7];
  const float* W_temp_ui = (const float*)d_in[8];
  const float* W_val_ui  = (const float*)d_in[9];
  const float* att_ui    = (const float*)d_in[10];
  const float* W_src_iu  = (const float*)d_in[11];
  const float* W_dst_iu  = (const float*)d_in[12];
  const float* W_temp_iu = (const float*)d_in[13];
  const float* W_val_iu  = (const float*)d_in[14];
  const float* att_iu    = (const float*)d_in[15];
  const float* Wout_user = (const float*)d_in[16];
  const float* bout_user = (const float*)d_in[17];
  const float* Wout_item = (const float*)d_in[18];
  const float* bout_item = (const float*)d_in[19];
  const float* lnw_user  = (const float*)d_in[20];
  const float* lnb_user  = (const float*)d_in[21];
  const float* lnw_item  = (const float*)d_in[22];
  const float* lnb_item  = (const float*)d_in[23];
  float* out = (float*)d_out;
  float* ws  = (float*)d_ws;

  // workspace layout: 5*N*128 + 4*N*4 + E*4 floats ~= 139 MB
  float* agg_item = ws;
  float* agg_user = agg_item + (size_t)NN * HIDD;
  float* h_s      = agg_user + (size_t)NN * HIDD;
  float* h_d      = h_s + (size_t)NN * HIDD;
  float* v_s      = h_d + (size_t)NN * HIDD;
  float* score_s  = v_s + (size_t)NN * HIDD;
  float* score_d  = score_s + (size_t)NN * HH;
  int*   mkey     = (int*)(score_d + (size_t)NN * HH);
  float* ssum     = (float*)(mkey + (size_t)NN * HH);
  float* exbuf    = ssum + (size_t)NN * HH;

  // zero both aggregation buffers (contiguous)
  int aggN = 2 * NN * HIDD;
  fill_f32_kernel<<<(aggN + 255) / 256, 256, 0, stream>>>(agg_item, 0.f, aggN);

  // direction user -> item (produces agg_item)
  run_direction(h_user, h_item, ei_ui, t_ui, W_src_ui, W_dst_ui, W_temp_ui,
                W_val_ui, att_ui, agg_item, h_s, h_d, v_s, score_s, score_d,
                mkey, ssum, exbuf, stream);
  // direction item -> user (produces agg_user)
  run_direction(h_item, h_user, ei_iu, t_iu, W_src_iu, W_dst_iu, W_temp_iu,
                W_val_iu, att_iu, agg_user, h_s, h_d, v_s, score_s, score_d,
                mkey, ssum, exbuf, stream);

  // output projections + residual + LayerNorm
  const int gemmBlocks = (NN / 16 + 7) / 8;  // 391
  const int lnBlocks = NN * 32 / 256;        // 6250 (exact)
  gemm128_wmma_kernel<<<gemmBlocks, 256, 0, stream>>>(agg_user, Wout_user, h_s, NN);
  ln_kernel<<<lnBlocks, 256, 0, stream>>>(h_user, h_s, bout_user, lnw_user, lnb_user,
                                          out, NN);
  gemm128_wmma_kernel<<<gemmBlocks, 256, 0, stream>>>(agg_item, Wout_item, h_d, NN);
  ln_kernel<<<lnBlocks, 256, 0, stream>>>(h_item, h_d, bout_item, lnw_item, lnb_item,
                                          out + (size_t)NN * HIDD, NN);
}